// GraphSageLayer_41128606826591
// MI455X (gfx1250) — compile-verified
//
#include <hip/hip_runtime.h>
#include <math.h>

typedef __attribute__((ext_vector_type(2))) float v2f;
typedef __attribute__((ext_vector_type(8))) float v8f;

#define DF 128          // feature dim
#define KDIM 256        // 2*D (concat width)
#define APITCH 260      // LDS pitch for A tile (bank-conflict-free: 260 % 64 == 4)
#define OPITCH 132      // LDS pitch for out tile

// ---------------------------------------------------------------------------
// Kernel 1: edge aggregation. One 32-thread group per edge; each lane moves
// 4 floats (float4 gather from h[src], 4x global_atomic_add_f32 into msg[dst]).
// Lane 0 bumps the degree counter. All atomics resolve in L2 (both arrays fit).
// ---------------------------------------------------------------------------
__global__ __launch_bounds__(256) void sage_aggregate(
    const float* __restrict__ h, const int* __restrict__ src,
    const int* __restrict__ dst, float* __restrict__ msg,
    float* __restrict__ deg, int E) {
  long long t = (long long)blockIdx.x * blockDim.x + threadIdx.x;
  int e = (int)(t >> 5);
  int lane = (int)(t & 31);
  if (e >= E) return;
  int s = src[e];
  int d = dst[e];
  if (lane == 0) atomicAdd(&deg[d], 1.0f);
  const float4 v = *(const float4*)(h + (size_t)s * DF + lane * 4);
  float* mrow = msg + (size_t)d * DF + lane * 4;
  atomicAdd(mrow + 0, v.x);
  atomicAdd(mrow + 1, v.y);
  atomicAdd(mrow + 2, v.z);
  atomicAdd(mrow + 3, v.w);
}

// ---------------------------------------------------------------------------
// Kernel 2: per-16-node tile: bundle = [h | msg/deg] (16x256) staged in LDS,
// then 8 waves each compute one 16x16 output tile with v_wmma_f32_16x16x4_f32
// over K=256 (64 WMMAs/wave), bias add, cross-wave L2 row norm, relu, snorm,
// residual.
// ---------------------------------------------------------------------------
__global__ __launch_bounds__(256) void sage_node_apply(
    const float* __restrict__ h, const float* __restrict__ snorm,
    const float* __restrict__ W, const float* __restrict__ bias,
    const float* __restrict__ msg, const float* __restrict__ deg,
    float* __restrict__ out, int N) {
  __shared__ float lds_a[16 * APITCH];   // bundle tile, padded
  __shared__ float lds_o[16 * OPITCH];   // output tile, padded
  __shared__ float lds_nrm[16];

  const int tid = threadIdx.x;
  const int base = blockIdx.x * 16;

  // ---- stage A tile: rows = 16 nodes, cols 0..127 = h, cols 128..255 = c ----
  for (int idx4 = tid; idx4 < 512; idx4 += 256) {    // 512 float4 = 16x128
    int idx = idx4 * 4;
    int r = idx >> 7;            // node row 0..15
    int c = idx & 127;           // feature col, multiple of 4
    int node = base + r; if (node >= N) node = N - 1;
    float4 hv = *(const float4*)(h + (size_t)node * DF + c);
    *(float4*)(&lds_a[r * APITCH + c]) = hv;
    float4 mv = *(const float4*)(msg + (size_t)node * DF + c);
    float inv = 1.0f / fmaxf(deg[node], 1.0f);
    float4 cv = make_float4(mv.x * inv, mv.y * inv, mv.z * inv, mv.w * inv);
    *(float4*)(&lds_a[r * APITCH + DF + c]) = cv;
  }
  __syncthreads();

  // ---- WMMA: wave w owns output columns [16w, 16w+16) ----
  const int wave = tid >> 5;
  const int lane = tid & 31;
  const int m    = lane & 15;              // A row (M) / B col (N) sub-index
  const int koff = (lane >> 4) << 1;       // half-wave K offset: 0 or 2
  const int ncol = wave * 16 + m;          // global output column

  const float* arow = &lds_a[m * APITCH + koff];          // A[m][k0+koff..+1]
  const float* brow = W + (size_t)ncol * KDIM + koff;     // B[k][n] = W[n][k]

  v8f acc = {};
#pragma unroll 8
  for (int k0 = 0; k0 < KDIM; k0 += 4) {
    v2f a = *(const v2f*)(arow + k0);
    v2f b = *(const v2f*)(brow + k0);
    acc = __builtin_amdgcn_wmma_f32_16x16x4_f32(
        /*neg_a=*/false, a, /*neg_b=*/false, b,
        /*c_mod=*/(short)0, acc, /*reuse_a=*/false, /*reuse_b=*/false);
  }

  // ---- spill tile to LDS with bias (D layout: row = v + 8*(lane/16)) ----
  const float bv = bias[ncol];
  const int mrow0 = (lane >> 4) * 8;
#pragma unroll
  for (int v = 0; v < 8; ++v) {
    lds_o[(mrow0 + v) * OPITCH + ncol] = acc[v] + bv;
  }
  __syncthreads();

  // ---- row-wise L2 norm over the full 128 columns ----
  if (tid < 16) {
    float s = 0.0f;
    const float* row = &lds_o[tid * OPITCH];
#pragma unroll 4
    for (int c = 0; c < DF; ++c) { float x = row[c]; s += x * x; }
    lds_nrm[tid] = fmaxf(sqrtf(s), 1e-12f);
  }
  __syncthreads();

  // ---- normalize, relu, graph-norm scale, residual, store ----
  for (int i = tid; i < 16 * DF; i += 256) {
    int r = i >> 7;
    int c = i & 127;
    int node = base + r;
    if (node >= N) continue;
    float val = lds_o[r * OPITCH + c] / lds_nrm[r];
    val = fmaxf(val, 0.0f) * snorm[node];
    out[(size_t)node * DF + c] = val + h[(size_t)node * DF + c];
  }
}

// ---------------------------------------------------------------------------
extern "C" void kernel_launch(void* const* d_in, const int* in_sizes, int n_in,
                              void* d_out, int out_size, void* d_ws, size_t ws_size,
                              hipStream_t stream) {
  const float* h     = (const float*)d_in[0];
  const float* snorm = (const float*)d_in[1];
  const float* W     = (const float*)d_in[2];
  const float* bias  = (const float*)d_in[3];
  const int*   src   = (const int*)d_in[4];
  const int*   dst   = (const int*)d_in[5];

  const int N = in_sizes[1];      // snorm_n has N elements
  const int E = in_sizes[4];      // src has E elements

  float* msg = (float*)d_ws;                       // [N, 128]
  float* deg = msg + (size_t)N * DF;               // [N]
  size_t zero_bytes = ((size_t)N * DF + (size_t)N) * sizeof(float);
  hipMemsetAsync(d_ws, 0, zero_bytes, stream);

  {
    long long threads = (long long)E * 32;
    int blocks = (int)((threads + 255) / 256);
    sage_aggregate<<<blocks, 256, 0, stream>>>(h, src, dst, msg, deg, E);
  }
  {
    int blocks = (N + 15) / 16;
    sage_node_apply<<<blocks, 256, 0, stream>>>(h, snorm, W, bias, msg, deg,
                                                (float*)d_out, N);
  }
}